// LinearNormalized_62672162783925
// MI455X (gfx1250) — compile-verified
//
#include <hip/hip_runtime.h>

// Row-wise L2 normalization of an (N, 4) fp32 tensor.
// Memory-bound: 256 MiB total traffic -> ~11.5 us roofline at 23.3 TB/s HBM.
//
// Main kernel is branch-free: launch covers exactly nrows/1024 full tiles, so
// each thread issues 4 unguarded non-temporal global_load_b128 (coalesced at
// blockDim stride, 4 loads in flight per thread), computes, and stores 4
// non-temporal global_store_b128. A separate guarded tail kernel handles any
// remainder rows (none for the 8388608-row reference input).

typedef float v4f __attribute__((ext_vector_type(4)));

#define ROWS_PER_THREAD 4
#define BLOCK_THREADS 256
#define ROWS_PER_BLOCK (BLOCK_THREADS * ROWS_PER_THREAD)

__device__ __forceinline__ v4f normalize_row(v4f v) {
    float n2 = fmaf(v.x, v.x, fmaf(v.y, v.y, fmaf(v.z, v.z, v.w * v.w)));
    float norm = sqrtf(n2);
    // Reference zero-guard: rows with zero norm divided by 1.0 (unchanged).
    float denom = (norm == 0.0f) ? 1.0f : norm;
    float inv = 1.0f / denom;   // one IEEE divide per row, then 4 muls
    v4f o;
    o.x = v.x * inv;
    o.y = v.y * inv;
    o.z = v.z * inv;
    o.w = v.w * inv;
    return o;
}

// Full tiles only: no bounds checks, no exec-mask branches.
__global__ __launch_bounds__(BLOCK_THREADS) void
l2norm_main_kernel(const v4f* __restrict__ x, v4f* __restrict__ out) {
    const int base = (int)blockIdx.x * ROWS_PER_BLOCK + (int)threadIdx.x;

    v4f v[ROWS_PER_THREAD];
#pragma unroll
    for (int k = 0; k < ROWS_PER_THREAD; ++k) {
        v[k] = __builtin_nontemporal_load(x + base + k * BLOCK_THREADS);
    }
#pragma unroll
    for (int k = 0; k < ROWS_PER_THREAD; ++k) {
        __builtin_nontemporal_store(normalize_row(v[k]), out + base + k * BLOCK_THREADS);
    }
}

// Remainder rows (row_offset .. nrows), one guarded row per thread.
__global__ __launch_bounds__(BLOCK_THREADS) void
l2norm_tail_kernel(const v4f* __restrict__ x, v4f* __restrict__ out,
                   int row_offset, int nrows) {
    const int r = row_offset + (int)blockIdx.x * BLOCK_THREADS + (int)threadIdx.x;
    if (r < nrows) {
        v4f v = __builtin_nontemporal_load(x + r);
        __builtin_nontemporal_store(normalize_row(v), out + r);
    }
}

extern "C" void kernel_launch(void* const* d_in, const int* in_sizes, int n_in,
                              void* d_out, int out_size, void* d_ws, size_t ws_size,
                              hipStream_t stream) {
    (void)n_in; (void)out_size; (void)d_ws; (void)ws_size;

    const v4f* x = (const v4f*)d_in[0];
    v4f* out = (v4f*)d_out;

    const int nrows = in_sizes[0] / 4;          // (8388608, 4) -> 8388608 rows
    const int full_blocks = nrows / ROWS_PER_BLOCK;
    const int covered = full_blocks * ROWS_PER_BLOCK;
    const int rem = nrows - covered;

    if (full_blocks > 0) {
        l2norm_main_kernel<<<full_blocks, BLOCK_THREADS, 0, stream>>>(x, out);
    }
    if (rem > 0) {
        const int tail_blocks = (rem + BLOCK_THREADS - 1) / BLOCK_THREADS;
        l2norm_tail_kernel<<<tail_blocks, BLOCK_THREADS, 0, stream>>>(x, out, covered, nrows);
    }
}